// GraphSAGEEncoder_1142461300900
// MI455X (gfx1250) — compile-verified
//
#include <hip/hip_runtime.h>

#define N_NODES 50000
#define N_EDGES 800000
#define DIM     128
#define EPSF    1e-5f

typedef __attribute__((ext_vector_type(2))) float v2f;
typedef __attribute__((ext_vector_type(8))) float v8f;

// ---------------------------------------------------------------------------
// Zero fill (grid-stride)
// ---------------------------------------------------------------------------
__global__ void fill_zero_kernel(float* __restrict__ p, int n) {
    int i = blockIdx.x * blockDim.x + threadIdx.x;
    int stride = gridDim.x * blockDim.x;
    for (; i < n; i += stride) p[i] = 0.0f;
}

// ---------------------------------------------------------------------------
// deg[dst] += 1 per edge
// ---------------------------------------------------------------------------
__global__ void degree_kernel(const int* __restrict__ ei, float* __restrict__ deg) {
    int e = blockIdx.x * blockDim.x + threadIdx.x;
    if (e < N_EDGES) {
        int dst = ei[N_EDGES + e];
        atomicAdd(&deg[dst], 1.0f);
    }
}

// ---------------------------------------------------------------------------
// agg[dst,:] += h[src,:]   one thread per (edge, 4-feature chunk)
// ---------------------------------------------------------------------------
__global__ void scatter_kernel(const float* __restrict__ h,
                               const int* __restrict__ ei,
                               float* __restrict__ agg) {
    long tid = (long)blockIdx.x * blockDim.x + threadIdx.x;
    const long total = (long)N_EDGES * (DIM / 4);
    if (tid >= total) return;
    int e = (int)(tid >> 5);   // DIM/4 == 32 chunks per edge
    int c = (int)(tid & 31);
    int src = ei[e];
    int dst = ei[N_EDGES + e];
    const float4 v = *(const float4*)(h + (long)src * DIM + c * 4);
    float* a = agg + (long)dst * DIM + c * 4;
    atomicAdd(a + 0, v.x);
    atomicAdd(a + 1, v.y);
    atomicAdd(a + 2, v.z);
    atomicAdd(a + 3, v.w);
}

// ---------------------------------------------------------------------------
// agg[i] /= max(deg[i/128], 1)
// ---------------------------------------------------------------------------
__global__ void mean_kernel(float* __restrict__ agg, const float* __restrict__ deg, int n) {
    int i = blockIdx.x * blockDim.x + threadIdx.x;
    int stride = gridDim.x * blockDim.x;
    for (; i < n; i += stride) {
        agg[i] /= fmaxf(deg[i >> 7], 1.0f);
    }
}

// ---------------------------------------------------------------------------
// out = LayerNorm(ReLU(mean @ Wl^T + b + h @ Wr^T)) * gamma + beta
// Block: 256 threads (8 waves), handles 16 rows x 128 cols.
// Wave w computes the 16x16 tile for cols [16w, 16w+16) via
// V_WMMA_F32_16X16X4_F32 over K=128 (32 steps, 2 WMMAs/step).
//
// WMMA f32 16x16x4 operand layouts (ISA 7.12.2):
//   A (16x4): lanes 0-15: M=lane, VGPR0/1 = K=0/1; lanes 16-31: K=2/3.
//   B (4x16): lanes 0-15: N=lane, VGPR0/1 = K=0/1; lanes 16-31: K=2/3.
//   C (16x16): VGPR r, lane l -> M = r + 8*(l>=16), N = l%16.
// B[k][n] = W[(j0+n)*DIM + k]  (W row-major [out,in], so B = W^T column view)
// ---------------------------------------------------------------------------
__global__ __launch_bounds__(256)
void sage_gemm_ln_kernel(const float* __restrict__ mean,
                         const float* __restrict__ h,
                         const float* __restrict__ Wl,
                         const float* __restrict__ bias,
                         const float* __restrict__ Wr,
                         const float* __restrict__ gam,
                         const float* __restrict__ bet,
                         float* __restrict__ out) {
    __shared__ float tile[16 * DIM];

    const int tid  = threadIdx.x;
    const int lane = tid & 31;
    const int wave = tid >> 5;            // 0..7
    const int row0 = blockIdx.x * 16;     // 50000/16 = 3125 blocks
    const int j0   = wave * 16;

    const int m16   = lane & 15;
    const int khalf = (lane >> 4) * 2;    // 0 for lanes 0-15, 2 for 16-31

    const float* arow_l = mean + (long)(row0 + m16) * DIM;
    const float* arow_r = h    + (long)(row0 + m16) * DIM;
    const float* brow_l = Wl   + (long)(j0 + m16) * DIM;
    const float* brow_r = Wr   + (long)(j0 + m16) * DIM;

    v8f c = {};
#pragma unroll 4
    for (int k0 = 0; k0 < DIM; k0 += 4) {
        v2f a_l = *(const v2f*)(arow_l + k0 + khalf);
        v2f b_l = *(const v2f*)(brow_l + k0 + khalf);
        v2f a_r = *(const v2f*)(arow_r + k0 + khalf);
        v2f b_r = *(const v2f*)(brow_r + k0 + khalf);
        c = __builtin_amdgcn_wmma_f32_16x16x4_f32(
                false, a_l, false, b_l, (short)0, c, false, false);
        c = __builtin_amdgcn_wmma_f32_16x16x4_f32(
                false, a_r, false, b_r, (short)0, c, false, false);
    }

    // Spill C tile to LDS in natural [row][col] layout.
    {
        const int n     = lane & 15;
        const int mbase = (lane >> 4) * 8;
#pragma unroll
        for (int r = 0; r < 8; ++r) {
            tile[(mbase + r) * DIM + j0 + n] = c[r];
        }
    }
    __syncthreads();

    // LayerNorm: row = tid/16 (16 rows), 16 threads/row, 8 cols/thread.
    const int row = tid >> 4;
    const int col = (tid & 15) * 8;

    float v[8];
    float s = 0.0f, sq = 0.0f;
#pragma unroll
    for (int i = 0; i < 8; ++i) {
        float x = tile[row * DIM + col + i] + bias[col + i];
        x = fmaxf(x, 0.0f);               // ReLU
        v[i] = x;
        s  += x;
        sq += x * x;
    }
    // Reduce across the 16 lanes of this row (masks < 16 stay in-half).
#pragma unroll
    for (int msk = 8; msk >= 1; msk >>= 1) {
        s  += __shfl_xor(s,  msk, 32);
        sq += __shfl_xor(sq, msk, 32);
    }
    const float mu   = s * (1.0f / DIM);
    const float var  = sq * (1.0f / DIM) - mu * mu;
    const float rstd = rsqrtf(var + EPSF);

    float* orow = out + (long)(row0 + row) * DIM + col;
#pragma unroll
    for (int i = 0; i < 8; ++i) {
        orow[i] = (v[i] - mu) * rstd * gam[col + i] + bet[col + i];
    }
}

// ---------------------------------------------------------------------------
// Launch: deg once; per layer: zero agg -> scatter -> mean -> gemm+LN
// ---------------------------------------------------------------------------
extern "C" void kernel_launch(void* const* d_in, const int* in_sizes, int n_in,
                              void* d_out, int out_size, void* d_ws, size_t ws_size,
                              hipStream_t stream) {
    const float* x     = (const float*)d_in[0];
    const int*   ei    = (const int*)  d_in[1];
    const float* Wl    = (const float*)d_in[2];
    const float* bl    = (const float*)d_in[3];
    const float* Wr    = (const float*)d_in[4];
    const float* gamma = (const float*)d_in[5];
    const float* beta  = (const float*)d_in[6];
    float*       out   = (float*)d_out;

    float* deg = (float*)d_ws;                       // N floats
    float* agg = deg + N_NODES;                      // N*D floats
    float* h1  = agg + (long)N_NODES * DIM;          // N*D floats

    // Degree (same for both layers).
    fill_zero_kernel<<<256, 256, 0, stream>>>(deg, N_NODES);
    degree_kernel<<<(N_EDGES + 255) / 256, 256, 0, stream>>>(ei, deg);

    const float* hin = x;
    for (int l = 0; l < 2; ++l) {
        float* hout = (l == 0) ? h1 : out;

        fill_zero_kernel<<<4096, 256, 0, stream>>>(agg, N_NODES * DIM);

        const long nscatter = (long)N_EDGES * (DIM / 4);
        scatter_kernel<<<(int)((nscatter + 255) / 256), 256, 0, stream>>>(hin, ei, agg);

        mean_kernel<<<4096, 256, 0, stream>>>(agg, deg, N_NODES * DIM);

        sage_gemm_ln_kernel<<<N_NODES / 16, 256, 0, stream>>>(
            agg, hin,
            Wl + (long)l * DIM * DIM, bl + (long)l * DIM,
            Wr + (long)l * DIM * DIM,
            gamma + (long)l * DIM, beta + (long)l * DIM,
            hout);

        hin = hout;
    }
}